// GatedMatchRNN_65094524338719
// MI455X (gfx1250) — compile-verified
//
#include <hip/hip_runtime.h>

#define Bc    32
#define L1c   128
#define L2c   512
#define Hc    256
#define GROUP 4   // batch rows per persistent block (32/4 = 8 blocks)

typedef float v2f __attribute__((ext_vector_type(2)));
typedef float v8f __attribute__((ext_vector_type(8)));

// D = A(16x4,f32) * B(4x16,f32) + C(16x16,f32)  -> v_wmma_f32_16x16x4_f32
__device__ __forceinline__ v8f wmma4(v2f a, v2f b, v8f c) {
  return __builtin_amdgcn_wmma_f32_16x16x4_f32(false, a, false, b, (short)0, c,
                                               false, false);
}

__device__ __forceinline__ float fast_rcp(float x) {
#if __has_builtin(__builtin_amdgcn_rcpf)
  return __builtin_amdgcn_rcpf(x);
#else
  return 1.0f / x;
#endif
}

__device__ __forceinline__ float fast_tanh(float x) {
#if __has_builtin(__builtin_amdgcn_tanhf)
  return __builtin_amdgcn_tanhf(x);          // native v_tanh_f32
#else
  float e = __expf(2.0f * x);                // native v_exp_f32 path
  return 1.0f - 2.0f * fast_rcp(e + 1.0f);   // exact limits at +-inf
#endif
}

__device__ __forceinline__ float fast_sigmoid(float x) {
  return fast_rcp(1.0f + __expf(-x));
}

// ---------------------------------------------------------------------------
// Kernel 1: hoisted projections  y_proj = y*Wq^T + bq,  x_up = x*Wup^T + bup
// Full 16x16 fp32 WMMA tiles, K accumulated in steps of 4.
// Grid: (B*L2 + B*L1)/16 M-tiles, 256 threads (8 waves, 2 N-tiles each).
// ---------------------------------------------------------------------------
__global__ __launch_bounds__(256) void precompute_proj(
    const float* __restrict__ y, const float* __restrict__ x,
    const float* __restrict__ Wq, const float* __restrict__ bq,
    const float* __restrict__ Wup, const float* __restrict__ bup,
    float* __restrict__ y_proj, float* __restrict__ x_up) {
  const int YT = (Bc * L2c) / 16;  // 1024 M-tiles for y_proj
  int blk = blockIdx.x;
  const float *src, *W, *bias;
  float* dst;
  int rowbase;
  if (blk < YT) {
    src = y;  W = Wq;  bias = bq;  dst = y_proj;  rowbase = blk * 16;
  } else {
    src = x;  W = Wup; bias = bup; dst = x_up;    rowbase = (blk - YT) * 16;
  }
  int wave = threadIdx.x >> 5;   // 0..7
  int lane = threadIdx.x & 31;
  int hf   = lane >> 4;          // K-half selector
  int r16  = lane & 15;          // A-row / B-col / D-col

  const float* arow = src + (size_t)(rowbase + r16) * Hc;
  for (int it = 0; it < 2; ++it) {
    int n0 = (wave * 2 + it) * 16;
    const float* brow = W + (size_t)(n0 + r16) * Hc;
    v8f acc = {0.f, 0.f, 0.f, 0.f, 0.f, 0.f, 0.f, 0.f};
    for (int k0 = 0; k0 < Hc; k0 += 4) {
      int kk = k0 + 2 * hf;
      v2f a = *(const v2f*)(arow + kk);
      v2f b = *(const v2f*)(brow + kk);
      acc = wmma4(a, b, acc);
    }
    float bv = bias[n0 + r16];
    float* dcol = dst + (size_t)(rowbase + 8 * hf) * Hc + n0 + r16;
#pragma unroll
    for (int i = 0; i < 8; ++i) dcol[(size_t)i * Hc] = acc[i] + bv;
  }
}

// ---------------------------------------------------------------------------
// Kernel 2: persistent scan. One block = GROUP batch rows, 1024 threads
// (32 wave32). All recurrent state in LDS; gate GEMMs batched across the
// GROUP rows as zero-padded M=16 WMMA tiles; scores/softmax/ct on VALU with
// native transcendentals. No inter-block sync needed (batch independent).
// ---------------------------------------------------------------------------
__global__ __launch_bounds__(1024) void match_scan(
    const float* __restrict__ x, const int* __restrict__ x_mask,
    const float* __restrict__ y, const int* __restrict__ y_mask,
    const float* __restrict__ h0, const float* __restrict__ c0,
    const float* __restrict__ Wvp, const float* __restrict__ bvp,
    const float* __restrict__ Vw, const float* __restrict__ Vb,
    const float* __restrict__ Wg, const float* __restrict__ bg,
    const float* __restrict__ Wih, const float* __restrict__ Whh,
    const float* __restrict__ bih, const float* __restrict__ bhh,
    const float* __restrict__ y_proj, const float* __restrict__ x_up,
    float* __restrict__ out) {
  __shared__ float sh_h[GROUP][Hc], sh_c[GROUP][Hc], sh_hid[GROUP][Hc];
  __shared__ float sh_xt[GROUP][Hc], sh_xup[GROUP][Hc];
  __shared__ float sh_li[GROUP][2 * Hc];
  __shared__ float poolA[GROUP * L2c];      // scores/alpha, then gates[n<512]
  __shared__ float poolB[GROUP * 2 * Hc];   // merge,        then gates[n>=512]
  __shared__ float sh_red[GROUP][8];
  __shared__ float sh_v[Hc];
  __shared__ int   sh_xm[GROUP];

  const int tid  = threadIdx.x;
  const int b0   = blockIdx.x * GROUP;
  const int wave = tid >> 5, lane = tid & 31;
  const int hf   = lane >> 4, r16 = lane & 15;
  const float vVb = Vb[0];

  {  // init carry + v vector
    int bb = tid >> 8, j = tid & 255;
    sh_h[bb][j] = h0[(b0 + bb) * Hc + j];
    sh_c[bb][j] = c0[(b0 + bb) * Hc + j];
    if (tid < Hc) sh_v[tid] = Vw[tid];
  }
  __syncthreads();

  for (int t = 0; t < L1c; ++t) {
    // ---- load x_t, xup_t, x-mask; prefetch next step's rows ----
    {
      int bb = tid >> 8, j = tid & 255;
      size_t row = (size_t)((b0 + bb) * L1c + t) * Hc;
      sh_xt[bb][j]  = x[row + j];
      sh_xup[bb][j] = x_up[row + j];
      if (j == 0) sh_xm[bb] = x_mask[(b0 + bb) * L1c + t];
      if (t + 1 < L1c) {  // global_prefetch_b8 into WGP$/L2
        __builtin_prefetch(x + row + Hc + j, 0, 0);
        __builtin_prefetch(x_up + row + Hc + j, 0, 0);
      }
    }
    __syncthreads();

    // ---- hid = h*Wvp^T + bvp  (WMMA, N=256 -> 16 tiles, waves 0..15) ----
    if (wave < 16) {
      int n0 = wave * 16;
      const float* brow = Wvp + (size_t)(n0 + r16) * Hc;
      v8f acc = {0.f, 0.f, 0.f, 0.f, 0.f, 0.f, 0.f, 0.f};
      for (int k0 = 0; k0 < Hc; k0 += 4) {
        int kk = k0 + 2 * hf;
        v2f a;
        a.x = (r16 < GROUP) ? sh_h[r16][kk] : 0.f;
        a.y = (r16 < GROUP) ? sh_h[r16][kk + 1] : 0.f;
        v2f b = *(const v2f*)(brow + kk);
        acc = wmma4(a, b, acc);
      }
      if (hf == 0) {
        float bv = bvp[n0 + r16];
#pragma unroll
        for (int i = 0; i < GROUP; ++i) sh_hid[i][n0 + r16] = acc[i] + bv;
      }
    } else {
      // idle waves warm the gate-weight rows they will consume this step
      // (overlaps L2 latency with the hid GEMM + upcoming score phase)
      const float* pg = Wg + (size_t)(wave * 16 + r16) * (2 * Hc);
      const float* p1 = Wih + (size_t)(wave * 16 + r16) * (2 * Hc);
      const float* p2 = Whh + (size_t)((wave + 32) * 16 + r16) * Hc;
#pragma unroll
      for (int o = 0; o < 2 * Hc; o += 128) {
        __builtin_prefetch(pg + o + hf * 64, 0, 0);
        __builtin_prefetch(p1 + o + hf * 64, 0, 0);
      }
#pragma unroll
      for (int o = 0; o < Hc; o += 128) __builtin_prefetch(p2 + o + hf * 64, 0, 0);
    }
    __syncthreads();

    // ---- scores: s[bb][l] = Vb + sum_h v[h]*tanh(xup+hid+y_proj) ----
#pragma unroll
    for (int rr = 0; rr < 2; ++rr) {
      int row = tid + rr * 1024;       // 0..2047 = GROUP*L2
      int bb = row >> 9, l = row & (L2c - 1);
      const float4* yp =
          (const float4*)(y_proj + (size_t)((b0 + bb) * L2c + l) * Hc);
      const float* hid = sh_hid[bb];
      const float* xup = sh_xup[bb];
      float acc0 = 0.f, acc1 = 0.f;
      for (int h4 = 0; h4 < Hc / 4; ++h4) {
        float4 q = yp[h4];
        int hh = h4 * 4;
        acc0 = fmaf(sh_v[hh + 0], fast_tanh(q.x + xup[hh + 0] + hid[hh + 0]), acc0);
        acc1 = fmaf(sh_v[hh + 1], fast_tanh(q.y + xup[hh + 1] + hid[hh + 1]), acc1);
        acc0 = fmaf(sh_v[hh + 2], fast_tanh(q.z + xup[hh + 2] + hid[hh + 2]), acc0);
        acc1 = fmaf(sh_v[hh + 3], fast_tanh(q.w + xup[hh + 3] + hid[hh + 3]), acc1);
      }
      float acc = acc0 + acc1 + vVb;
      if (y_mask[(b0 + bb) * L2c + l]) acc = -__builtin_inff();
      poolA[bb * L2c + l] = acc;
    }
    __syncthreads();

    // ---- masked softmax over L2 (per bb, 256 threads / bb) ----
    {
      int bb = tid >> 8, j = tid & 255, w8 = j >> 5;
      float s0 = poolA[bb * L2c + j], s1 = poolA[bb * L2c + j + 256];
      float m = fmaxf(s0, s1);
      for (int off = 16; off > 0; off >>= 1) m = fmaxf(m, __shfl_xor(m, off, 32));
      if (lane == 0) sh_red[bb][w8] = m;
      __syncthreads();
      float M = sh_red[bb][0];
#pragma unroll
      for (int w = 1; w < 8; ++w) M = fmaxf(M, sh_red[bb][w]);
      float e0 = __expf(s0 - M), e1 = __expf(s1 - M);
      float ssum = e0 + e1;
      for (int off = 16; off > 0; off >>= 1) ssum += __shfl_xor(ssum, off, 32);
      __syncthreads();  // everyone done reading max partials
      if (lane == 0) sh_red[bb][w8] = ssum;
      __syncthreads();
      float S = 0.f;
#pragma unroll
      for (int w = 0; w < 8; ++w) S += sh_red[bb][w];
      float inv = fast_rcp(S);
      poolA[bb * L2c + j]       = e0 * inv;
      poolA[bb * L2c + j + 256] = e1 * inv;
    }
    __syncthreads();

    // ---- ct = alpha * y ; merge = [x_t, ct] ----
    {
      int bb = tid >> 8, j = tid & 255;
      const float* yb = y + (size_t)(b0 + bb) * L2c * Hc + j;
      const float* al = &poolA[bb * L2c];
      float a0 = 0.f, a1 = 0.f, a2 = 0.f, a3 = 0.f;
      for (int l = 0; l < L2c; l += 4) {
        a0 = fmaf(al[l + 0], yb[(size_t)(l + 0) * Hc], a0);
        a1 = fmaf(al[l + 1], yb[(size_t)(l + 1) * Hc], a1);
        a2 = fmaf(al[l + 2], yb[(size_t)(l + 2) * Hc], a2);
        a3 = fmaf(al[l + 3], yb[(size_t)(l + 3) * Hc], a3);
      }
      float acc = (a0 + a1) + (a2 + a3);
      poolB[bb * (2 * Hc) + j]      = sh_xt[bb][j];
      poolB[bb * (2 * Hc) + Hc + j] = acc;
    }
    __syncthreads();

    // ---- gt = sigmoid(merge*Wg^T + bg); li = xm ? 0 : gt*merge  (WMMA) ----
    {
      int n0 = wave * 16;  // 32 waves cover N=512
      const float* brow = Wg + (size_t)(n0 + r16) * (2 * Hc);
      v8f acc = {0.f, 0.f, 0.f, 0.f, 0.f, 0.f, 0.f, 0.f};
      for (int k0 = 0; k0 < 2 * Hc; k0 += 4) {
        int kk = k0 + 2 * hf;
        v2f a;
        a.x = (r16 < GROUP) ? poolB[r16 * (2 * Hc) + kk] : 0.f;
        a.y = (r16 < GROUP) ? poolB[r16 * (2 * Hc) + kk + 1] : 0.f;
        v2f b = *(const v2f*)(brow + kk);
        acc = wmma4(a, b, acc);
      }
      if (hf == 0) {
        float bv = bg[n0 + r16];
#pragma unroll
        for (int i = 0; i < GROUP; ++i) {
          float g = fast_sigmoid(acc[i] + bv);
          sh_li[i][n0 + r16] =
              sh_xm[i] ? 0.f : g * poolB[i * (2 * Hc) + n0 + r16];
        }
      }
    }
    __syncthreads();

    // ---- gates = li*Wih^T + h*Whh^T + bih + bhh  (WMMA, N=1024) ----
    for (int it = 0; it < 2; ++it) {
      int n0 = (wave + it * 32) * 16;
      const float* brow1 = Wih + (size_t)(n0 + r16) * (2 * Hc);
      const float* brow2 = Whh + (size_t)(n0 + r16) * Hc;
      v8f acc = {0.f, 0.f, 0.f, 0.f, 0.f, 0.f, 0.f, 0.f};
      for (int k0 = 0; k0 < 2 * Hc; k0 += 4) {
        int kk = k0 + 2 * hf;
        v2f a;
        a.x = (r16 < GROUP) ? sh_li[r16][kk] : 0.f;
        a.y = (r16 < GROUP) ? sh_li[r16][kk + 1] : 0.f;
        v2f b = *(const v2f*)(brow1 + kk);
        acc = wmma4(a, b, acc);
      }
      for (int k0 = 0; k0 < Hc; k0 += 4) {
        int kk = k0 + 2 * hf;
        v2f a;
        a.x = (r16 < GROUP) ? sh_h[r16][kk] : 0.f;
        a.y = (r16 < GROUP) ? sh_h[r16][kk + 1] : 0.f;
        v2f b = *(const v2f*)(brow2 + kk);
        acc = wmma4(a, b, acc);
      }
      if (hf == 0) {
        float bv = bih[n0 + r16] + bhh[n0 + r16];
        if (it == 0) {  // n < 512 -> poolA
#pragma unroll
          for (int i = 0; i < GROUP; ++i)
            poolA[i * L2c + n0 + r16] = acc[i] + bv;
        } else {        // n >= 512 -> poolB
#pragma unroll
          for (int i = 0; i < GROUP; ++i)
            poolB[i * (2 * Hc) + (n0 - 512) + r16] = acc[i] + bv;
        }
      }
    }
    __syncthreads();

    // ---- LSTM cell (gate order i,f,g,o); write h_t (non-temporal) ----
    {
      int bb = tid >> 8, j = tid & 255;
      float gi = fast_sigmoid(poolA[bb * L2c + j]);
      float gf = fast_sigmoid(poolA[bb * L2c + 256 + j]);
      float gg = fast_tanh(poolB[bb * (2 * Hc) + j]);
      float go = fast_sigmoid(poolB[bb * (2 * Hc) + 256 + j]);
      float cn = gf * sh_c[bb][j] + gi * gg;
      float hn = go * fast_tanh(cn);
      hn = sh_xm[bb] ? 0.f : hn;
      sh_c[bb][j] = cn;
      sh_h[bb][j] = hn;
      __builtin_nontemporal_store(
          hn, &out[(size_t)((b0 + bb) * L1c + t) * Hc + j]);
    }
    __syncthreads();
  }
}

// ---------------------------------------------------------------------------
extern "C" void kernel_launch(void* const* d_in, const int* in_sizes, int n_in,
                              void* d_out, int out_size, void* d_ws,
                              size_t ws_size, hipStream_t stream) {
  const float* x   = (const float*)d_in[0];
  const int*   xm  = (const int*)d_in[1];
  const float* y   = (const float*)d_in[2];
  const int*   ym  = (const int*)d_in[3];
  const float* h0  = (const float*)d_in[4];
  const float* c0  = (const float*)d_in[5];
  const float* Wq  = (const float*)d_in[6];
  const float* bq  = (const float*)d_in[7];
  const float* Wup = (const float*)d_in[8];
  const float* bup = (const float*)d_in[9];
  const float* Wvp = (const float*)d_in[10];
  const float* bvp = (const float*)d_in[11];
  const float* Vw  = (const float*)d_in[12];
  const float* Vb  = (const float*)d_in[13];
  const float* Wg  = (const float*)d_in[14];
  const float* bg  = (const float*)d_in[15];
  const float* Wih = (const float*)d_in[16];
  const float* Whh = (const float*)d_in[17];
  const float* bih = (const float*)d_in[18];
  const float* bhh = (const float*)d_in[19];
  float* out = (float*)d_out;

  float* y_proj = (float*)d_ws;                       // [B,L2,H] f32, 16.8MB
  float* x_up   = y_proj + (size_t)Bc * L2c * Hc;     // [B,L1,H] f32,  4.2MB

  int nblk1 = (Bc * L2c + Bc * L1c) / 16;  // 1280 M-tiles
  precompute_proj<<<nblk1, 256, 0, stream>>>(y, x, Wq, bq, Wup, bup, y_proj,
                                             x_up);
  match_scan<<<Bc / GROUP, 1024, 0, stream>>>(x, xm, y, ym, h0, c0, Wvp, bvp,
                                              Vw, Vb, Wg, bg, Wih, Whh, bih,
                                              bhh, y_proj, x_up, out);
}